// FsCoding10_20684562497551
// MI455X (gfx1250) — compile-verified
//
#include <hip/hip_runtime.h>

// FsCoding10 fast-forward path, elementwise over N = 8*2048*4096 fp32 values.
//
// Simplification: (v - T)/(|v| + 1) > 0  <=>  v > T   (denominator strictly > 0),
// so the per-step divide is dead in the forward pass. Each of the 16 steps is
// 4 VALU ops: v_fmac (v -= z*h), v_cmp_gt, v_cndmask (z = 0/1), v_fmac (acc += z*d).
// Final sign(x) multiply == XOR of x's sign bit into acc (acc==0 when x==0 since
// all T[t] > 0, so the x==0 case is already exact).
//
// Memory plan: streaming 256 MB in + 256 MB out with no reuse -> 128-bit
// non-temporal global loads/stores (bypass-ish L2 residency), grid-stride loop,
// global_prefetch for the next stride tile. Coefficients (48 floats) come from
// uniform kernarg pointers -> promoted to SGPR s_loads, read once per thread.

typedef float vfloat4 __attribute__((ext_vector_type(4)));

__device__ __forceinline__ float fs_elem(float xin,
                                         const float* __restrict__ hc,
                                         const float* __restrict__ dc,
                                         const float* __restrict__ Tc) {
    float v   = __builtin_fabsf(xin);
    float z   = 0.0f;
    float acc = 0.0f;
#pragma unroll
    for (int t = 0; t < 16; ++t) {
        v   = __builtin_fmaf(z, -hc[t], v);     // v -= z * h[t]
        z   = (v > Tc[t]) ? 1.0f : 0.0f;        // spike: sign of (v - T[t])
        acc = __builtin_fmaf(z, dc[t], acc);    // out += z * d[t]
    }
    unsigned sbit = __float_as_uint(xin) & 0x80000000u;
    return __uint_as_float(__float_as_uint(acc) ^ sbit);   // acc * sign(x)
}

__global__ __launch_bounds__(256) void fs10_vec4_kernel(
    const float* __restrict__ x,
    const float* __restrict__ h,
    const float* __restrict__ d,
    const float* __restrict__ T,
    float* __restrict__ out,
    int n4) {
    // Coefficients: uniform addresses -> SGPRs after full unroll.
    float hc[16], dc[16], Tc[16];
#pragma unroll
    for (int t = 0; t < 16; ++t) {
        hc[t] = h[t];
        dc[t] = d[t];
        Tc[t] = T[t];
    }

    const vfloat4* __restrict__ x4 = reinterpret_cast<const vfloat4*>(x);
    vfloat4* __restrict__ o4       = reinterpret_cast<vfloat4*>(out);

    const int stride = (int)(gridDim.x * blockDim.x);
    for (int i = (int)(blockIdx.x * blockDim.x + threadIdx.x); i < n4; i += stride) {
        // Prefetch next grid-stride tile (gfx1250: global_prefetch_b8).
        if (i + stride < n4) {
            __builtin_prefetch((const void*)&x4[i + stride], 0, 0);
        }
        vfloat4 xv = __builtin_nontemporal_load(&x4[i]);   // global_load_b128 th:NT
        vfloat4 ov;
        ov.x = fs_elem(xv.x, hc, dc, Tc);
        ov.y = fs_elem(xv.y, hc, dc, Tc);
        ov.z = fs_elem(xv.z, hc, dc, Tc);
        ov.w = fs_elem(xv.w, hc, dc, Tc);
        __builtin_nontemporal_store(ov, &o4[i]);           // global_store_b128 th:NT
    }
}

// Scalar cleanup for N % 4 != 0 (not hit for 8*2048*4096, but kept for safety).
__global__ void fs10_tail_kernel(const float* __restrict__ x,
                                 const float* __restrict__ h,
                                 const float* __restrict__ d,
                                 const float* __restrict__ T,
                                 float* __restrict__ out,
                                 int first, int n) {
    float hc[16], dc[16], Tc[16];
#pragma unroll
    for (int t = 0; t < 16; ++t) {
        hc[t] = h[t];
        dc[t] = d[t];
        Tc[t] = T[t];
    }
    int i = first + (int)threadIdx.x;
    if (i < n) out[i] = fs_elem(x[i], hc, dc, Tc);
}

extern "C" void kernel_launch(void* const* d_in, const int* in_sizes, int n_in,
                              void* d_out, int out_size, void* d_ws, size_t ws_size,
                              hipStream_t stream) {
    const float* x = (const float*)d_in[0];
    const float* h = (const float*)d_in[1];
    const float* d = (const float*)d_in[2];
    const float* T = (const float*)d_in[3];
    float* out     = (float*)d_out;

    const int N   = in_sizes[0];      // 8*2048*4096 = 67,108,864
    const int n4  = N >> 2;           // 16,777,216 float4s
    const int rem = N & 3;

    const int threads = 256;          // 8 wave32 per block
    int blocks = (n4 + threads - 1) / threads;
    if (blocks > 32768) blocks = 32768;   // persistent grid-stride; ~2 float4/thread here
    if (blocks < 1) blocks = 1;

    fs10_vec4_kernel<<<blocks, threads, 0, stream>>>(x, h, d, T, out, n4);
    if (rem) {
        fs10_tail_kernel<<<1, 32, 0, stream>>>(x, h, d, T, out, n4 << 2, N);
    }
}